// PAM_9225589752594
// MI455X (gfx1250) — compile-verified
//
#include <hip/hip_runtime.h>

typedef __attribute__((ext_vector_type(2))) float v2f;
typedef __attribute__((ext_vector_type(8))) float v8f;

namespace {
constexpr int kB = 2;
constexpr int kN = 9216;                              // H*W
constexpr int kNT = kN / 16;                          // 576 16-wide column tiles
constexpr int kWavesPerBlock = 4;                     // 128 threads
constexpr int kBlocksPerBatch = kNT / kWavesPerBlock; // 144
}

// Kernel 1: channel softmax (C=2) -> v[b][n] = {v0, v1} interleaved (also = xs).
__global__ void pam_softmax_c(const float* __restrict__ x, float2* __restrict__ v) {
  int t = blockIdx.x * blockDim.x + threadIdx.x;
  if (t >= kB * kN) return;
  int b = t / kN, n = t - b * kN;
  const float* xb = x + (size_t)b * 2 * kN;
  float x0 = xb[n], x1 = xb[kN + n];
  float m = fmaxf(x0, x1);
  float e0 = __expf(x0 - m), e1 = __expf(x1 - m);
  float inv = 1.0f / (e0 + e1);
  float2 r;
  r.x = e0 * inv;
  r.y = e1 * inv;
  v[(size_t)b * kN + n] = r;
}

// Kernel 2: one wave handles 16 attention rows. energy tile = v_wmma_f32_16x16x4_f32
// (K=0,1 are the two channels; K=2,3 zero-padded -> lanes 16..31 of A/B frags are 0).
__global__ __launch_bounds__(kWavesPerBlock * 32)
void pam_attention(const float2* __restrict__ v,
                   const float* __restrict__ conv_w,
                   const float* __restrict__ conv_b,
                   const float* __restrict__ gamma,
                   float* __restrict__ yOut,
                   float* __restrict__ attOut) {
  extern __shared__ float2 lv[];           // kN float2 (73728 B dynamic LDS)

  const int b = blockIdx.x / kBlocksPerBatch;
  const int tile0 = (blockIdx.x % kBlocksPerBatch) * kWavesPerBlock;

  const float2* vb = v + (size_t)b * kN;
  for (int i = threadIdx.x; i < kN; i += blockDim.x) lv[i] = vb[i];
  __syncthreads();

  const int lane = threadIdx.x & 31;
  const int wave = threadIdx.x >> 5;
  const int laneLo = lane & 15;
  const int hi = lane >> 4;                // 0: rows r / K=0,1   1: rows r+8 / K=2,3(=0)
  const int nb = (tile0 + wave) * 16;      // first of this wave's 16 rows

  // A fragment: unconditional LDS load, then cndmask (no exec divergence).
  const float2 pa = lv[nb + laneLo];
  v2f afrag;
  afrag.x = hi ? 0.0f : pa.x;
  afrag.y = hi ? 0.0f : pa.y;

  // ---------- pass 1: per-row max of energy ----------
  float mx[8];
#pragma unroll
  for (int r = 0; r < 8; ++r) mx[r] = -3.402823466e38f;
  for (int mt = 0; mt < kNT; ++mt) {
    const float2 p = lv[mt * 16 + laneLo];           // one ds_load_b64
    v2f bfrag;
    bfrag.x = hi ? 0.0f : p.x;
    bfrag.y = hi ? 0.0f : p.y;
    v8f acc = {};
    acc = __builtin_amdgcn_wmma_f32_16x16x4_f32(false, afrag, false, bfrag,
                                                (short)0, acc, false, false);
#pragma unroll
    for (int r = 0; r < 8; ++r) mx[r] = fmaxf(mx[r], acc[r]);
  }
#pragma unroll
  for (int r = 0; r < 8; ++r) {
#pragma unroll
    for (int off = 1; off < 16; off <<= 1)
      mx[r] = fmaxf(mx[r], __shfl_xor(mx[r], off, 32));
  }

  // ---------- pass 2: per-row sum of exp(e - max) ----------
  float sum[8];
#pragma unroll
  for (int r = 0; r < 8; ++r) sum[r] = 0.0f;
  for (int mt = 0; mt < kNT; ++mt) {
    const float2 p = lv[mt * 16 + laneLo];
    v2f bfrag;
    bfrag.x = hi ? 0.0f : p.x;
    bfrag.y = hi ? 0.0f : p.y;
    v8f acc = {};
    acc = __builtin_amdgcn_wmma_f32_16x16x4_f32(false, afrag, false, bfrag,
                                                (short)0, acc, false, false);
#pragma unroll
    for (int r = 0; r < 8; ++r) sum[r] += __expf(acc[r] - mx[r]);
  }
  float rinv[8];
#pragma unroll
  for (int r = 0; r < 8; ++r) {
#pragma unroll
    for (int off = 1; off < 16; off <<= 1)
      sum[r] += __shfl_xor(sum[r], off, 32);
    rinv[r] = 1.0f / sum[r];
  }

  // ---------- pass 3: write attention (non-temporal) + out = att . v^T ----------
  float* attB = attOut + (size_t)b * kN * kN;
  float* rowPtr[8];
#pragma unroll
  for (int r = 0; r < 8; ++r)
    rowPtr[r] = attB + (size_t)(nb + r + 8 * hi) * kN + laneLo;

  float o0[8], o1[8];
#pragma unroll
  for (int r = 0; r < 8; ++r) { o0[r] = 0.0f; o1[r] = 0.0f; }

  for (int mt = 0; mt < kNT; ++mt) {
    const int mb = mt * 16;
    const float2 p = lv[mb + laneLo];                // one ds_load_b64, reused below
    v2f bfrag;
    bfrag.x = hi ? 0.0f : p.x;
    bfrag.y = hi ? 0.0f : p.y;
    v8f acc = {};
    acc = __builtin_amdgcn_wmma_f32_16x16x4_f32(false, afrag, false, bfrag,
                                                (short)0, acc, false, false);
#pragma unroll
    for (int r = 0; r < 8; ++r) {
      float att = __expf(acc[r] - mx[r]) * rinv[r];
      __builtin_nontemporal_store(att, rowPtr[r] + mb);  // stream past L2, never re-read
      o0[r] = fmaf(att, p.x, o0[r]);
      o1[r] = fmaf(att, p.y, o1[r]);
    }
  }

  // reduce out-accumulators across each 16-lane half
#pragma unroll
  for (int r = 0; r < 8; ++r) {
#pragma unroll
    for (int off = 1; off < 16; off <<= 1) {
      o0[r] += __shfl_xor(o0[r], off, 32);
      o1[r] += __shfl_xor(o1[r], off, 32);
    }
  }

  // epilogue: y = sigmoid(w0*(g*out0+xs0) + w1*(g*out1+xs1) + bias)
  const float g = gamma[0], w0 = conv_w[0], w1 = conv_w[1], cb = conv_b[0];
  if (laneLo == 0) {
#pragma unroll
    for (int r = 0; r < 8; ++r) {
      int n = nb + r + 8 * hi;
      float2 xs = lv[n];
      float pre0 = fmaf(g, o0[r], xs.x);
      float pre1 = fmaf(g, o1[r], xs.y);
      float y = fmaf(w0, pre0, fmaf(w1, pre1, cb));
      yOut[(size_t)b * kN + n] = 1.0f / (1.0f + __expf(-y));
    }
  }
}

extern "C" void kernel_launch(void* const* d_in, const int* in_sizes, int n_in,
                              void* d_out, int out_size, void* d_ws, size_t ws_size,
                              hipStream_t stream) {
  const float* x      = (const float*)d_in[0];
  const float* conv_w = (const float*)d_in[1];
  const float* conv_b = (const float*)d_in[2];
  const float* gamma  = (const float*)d_in[3];

  float* out    = (float*)d_out;
  float* yOut   = out;                       // [B,1,H,W] = 18432 floats
  float* attOut = out + (size_t)kB * kN;     // [B,N,N]
  float2* v     = (float2*)d_ws;             // [B,N] float2 = 147456 bytes

  int bn = kB * kN;
  pam_softmax_c<<<(bn + 255) / 256, 256, 0, stream>>>(x, v);

  size_t ldsBytes = (size_t)kN * sizeof(float2);  // 73728 B
  pam_attention<<<kB * kBlocksPerBatch, kWavesPerBlock * 32, ldsBytes, stream>>>(
      v, conv_w, conv_b, gamma, yOut, attOut);
}